// QLSTM_65481071397427
// MI455X (gfx1250) — compile-verified
//
#include <hip/hip_runtime.h>

// ---------------- problem constants (match reference) ----------------
#define T_LEN 256
#define B_SZ  4096
#define D_IN  32
#define NQ    8
#define FAN   (D_IN + NQ)   // 40

typedef float v2f __attribute__((ext_vector_type(2)));
typedef float v8f __attribute__((ext_vector_type(8)));

// ---------------------------------------------------------------------
// Prep: pack B-matrix (Wx^T in [n][k] layout, n = gate*8+q, k = 0..31)
// and fold bias+theta into BT[n]. One tiny block.
// ---------------------------------------------------------------------
__global__ void qlstm_prep(const float* Wf, const float* bf,
                           const float* Wi, const float* bi,
                           const float* Wu, const float* bu,
                           const float* Wo, const float* bo,
                           const float* thf, const float* thi,
                           const float* thu, const float* tho,
                           float* __restrict__ WB, float* __restrict__ BT) {
  const float* W[4]  = {Wf, Wi, Wu, Wo};
  const float* bb[4] = {bf, bi, bu, bo};
  const float* th[4] = {thf, thi, thu, tho};
  int idx = threadIdx.x;            // blockDim.x == 1024
  if (idx < 32 * 32) {
    int n = idx >> 5, k = idx & 31;        // WB[n][k] = W_all[n][k]
    WB[n * 32 + k] = W[n >> 3][(n & 7) * FAN + k];
  }
  if (idx < 32) {
    int g = idx >> 3, q = idx & 7;
    BT[idx] = bb[g][q] + th[g][q];
  }
}

// ---------------------------------------------------------------------
// Phase 1: ZX[r, n] = sum_k X[r,k] * W_all[n,k] + (bias+theta)[n]
// r = t*B + b  (M = T*B rows), K = 32, N = 32.
// One wave per 16-row tile; fp32 WMMA 16x16x4, 2 N-tiles x 8 K-steps.
// ---------------------------------------------------------------------
__global__ void qlstm_gemm(const float* __restrict__ X,
                           const float* __restrict__ WB,
                           const float* __restrict__ BT,
                           float* __restrict__ ZX) {
  const int  lane = threadIdx.x & 31;
  const int  wave = threadIdx.x >> 5;
  const long tile = (long)blockIdx.x * 8 + wave;   // blockDim.x == 256
  const long row0 = tile * 16;
  const int  M    = lane & 15;     // A row / D column index (lane%16)
  const int  hi   = lane >> 4;     // K-half select for A/B, M-half for D

  // A block 16x32: per lane 8 x v2f, lane holds row (row0+M),
  // K = 4*kk + 2*hi + {0,1}  (matches 32-bit A 16x4 VGPR layout)
  v2f a[8];
  const float* arow = X + (row0 + M) * 32;
#pragma unroll
  for (int kk = 0; kk < 8; ++kk)
    a[kk] = *(const v2f*)(arow + kk * 4 + hi * 2);

  // C init = bias+theta per output column n = ntile*16 + lane%16
  v8f c0, c1;
  float bt0 = BT[M];
  float bt1 = BT[16 + M];
#pragma unroll
  for (int v = 0; v < 8; ++v) { c0[v] = bt0; c1[v] = bt1; }

#pragma unroll
  for (int kk = 0; kk < 8; ++kk) {
    // B 4x16: lane holds col n = ntile*16 + M, K = 4*kk + 2*hi + {0,1}
    v2f b0 = *(const v2f*)(WB + M * 32        + kk * 4 + hi * 2);
    v2f b1 = *(const v2f*)(WB + (16 + M) * 32 + kk * 4 + hi * 2);
    c0 = __builtin_amdgcn_wmma_f32_16x16x4_f32(false, a[kk], false, b0,
                                               (short)0, c0, false, false);
    c1 = __builtin_amdgcn_wmma_f32_16x16x4_f32(false, a[kk], false, b1,
                                               (short)0, c1, false, false);
  }

  // D layout: VGPR v -> row M = v + 8*hi, col N = lane%16 (+16 for c1)
#pragma unroll
  for (int v = 0; v < 8; ++v) {
    long r = row0 + v + 8 * hi;
    ZX[r * 32 + M]      = c0[v];
    ZX[r * 32 + 16 + M] = c1[v];
  }
}

// ---------------------------------------------------------------------
// Phase 2: recurrence. 8 lanes per batch element (lane = qubit q).
// ---------------------------------------------------------------------
__device__ __forceinline__ float sigmoid_f(float x) {
  return 1.0f / (1.0f + __expf(-x));
}
__device__ __forceinline__ float tanh_f(float x) {
  float e = __expf(2.0f * fabsf(x));
  float r = 1.0f - 2.0f / (e + 1.0f);
  return copysignf(r, x);
}

__global__ void qlstm_rec(const float* __restrict__ ZX,
                          const float* __restrict__ Wf,
                          const float* __restrict__ Wi,
                          const float* __restrict__ Wu,
                          const float* __restrict__ Wo,
                          float* __restrict__ out) {
  const int id = blockIdx.x * blockDim.x + threadIdx.x;  // 0 .. B*8-1
  const int q  = id & 7;
  const int b  = id >> 3;

  // Recurrent weight rows for this lane's 4 gate outputs (cols 32..39)
  float whf[8], whi[8], whu[8], who[8];
#pragma unroll
  for (int j = 0; j < 8; ++j) {
    whf[j] = Wf[q * FAN + 32 + j];
    whi[j] = Wi[q * FAN + 32 + j];
    whu[j] = Wu[q * FAN + 32 + j];
    who[j] = Wo[q * FAN + 32 + j];
  }

  float h[8];
#pragma unroll
  for (int j = 0; j < 8; ++j) h[j] = 0.0f;
  float cst = 0.0f;
  float ht  = 0.0f;

  for (int t = 0; t < T_LEN; ++t) {
    const float* zr = ZX + ((long)t * B_SZ + b) * 32;
    float zf = zr[q], zi = zr[8 + q], zu = zr[16 + q], zo = zr[24 + q];
#pragma unroll
    for (int j = 0; j < 8; ++j) {
      zf = fmaf(h[j], whf[j], zf);
      zi = fmaf(h[j], whi[j], zi);
      zu = fmaf(h[j], whu[j], zu);
      zo = fmaf(h[j], who[j], zo);
    }
    float af = __cosf(zf), ai = __cosf(zi), au = __cosf(zu), ao = __cosf(zo);

    // inclusive multiplicative scan over the 8-lane qubit group (cumprod)
#pragma unroll
    for (int d = 1; d < 8; d <<= 1) {
      float tf = __shfl_up(af, (unsigned)d, 8);
      float ti = __shfl_up(ai, (unsigned)d, 8);
      float tu = __shfl_up(au, (unsigned)d, 8);
      float to = __shfl_up(ao, (unsigned)d, 8);
      if (q >= d) { af *= tf; ai *= ti; au *= tu; ao *= to; }
    }

    float fg = sigmoid_f(af);
    float ig = sigmoid_f(ai);
    float gg = tanh_f(au);
    float og = sigmoid_f(ao);
    cst = fg * cst + ig * gg;
    ht  = og * tanh_f(cst);

#pragma unroll
    for (int j = 0; j < 8; ++j) h[j] = __shfl(ht, j, 8);

    out[((long)t * B_SZ + b) * 8 + q] = ht;   // outputs[t][b][q]
  }

  // hx then cx appended after outputs
  const long base = (long)T_LEN * B_SZ * 8;
  out[base + id] = ht;
  out[base + (long)B_SZ * 8 + id] = cst;
}

// ---------------------------------------------------------------------
extern "C" void kernel_launch(void* const* d_in, const int* in_sizes, int n_in,
                              void* d_out, int out_size, void* d_ws, size_t ws_size,
                              hipStream_t stream) {
  (void)in_sizes; (void)n_in; (void)out_size; (void)ws_size;
  const float* X   = (const float*)d_in[0];
  const float* Wf  = (const float*)d_in[1];
  const float* bf  = (const float*)d_in[2];
  const float* Wi  = (const float*)d_in[3];
  const float* bi  = (const float*)d_in[4];
  const float* Wu  = (const float*)d_in[5];
  const float* bu  = (const float*)d_in[6];
  const float* Wo  = (const float*)d_in[7];
  const float* bo  = (const float*)d_in[8];
  const float* thf = (const float*)d_in[9];
  const float* thi = (const float*)d_in[10];
  const float* thu = (const float*)d_in[11];
  const float* tho = (const float*)d_in[12];

  // workspace layout: WB (32*32 f32) | BT (32 f32) | pad | ZX (T*B*32 f32)
  float* WB = (float*)d_ws;
  float* BT = WB + 32 * 32;
  float* ZX = (float*)((char*)d_ws + 8192);

  qlstm_prep<<<1, 1024, 0, stream>>>(Wf, bf, Wi, bi, Wu, bu, Wo, bo,
                                     thf, thi, thu, tho, WB, BT);

  // (T*B)/16 = 65536 wave tiles, 8 waves per 256-thread block
  qlstm_gemm<<<65536 / 8, 256, 0, stream>>>(X, WB, BT, ZX);

  // B*8 = 32768 lanes; small blocks to spread waves across WGPs
  qlstm_rec<<<(B_SZ * 8) / 64, 64, 0, stream>>>(ZX, Wf, Wi, Wu, Wo,
                                                (float*)d_out);
}